// GCPNetUpdate_49727131353246
// MI455X (gfx1250) — compile-verified
//
#include <hip/hip_runtime.h>
#include <hip/hip_bf16.h>

#define D 128
#define NN 20000
#define NE 200000
#define NA 600000

typedef float v2f __attribute__((ext_vector_type(2)));
typedef float v8f __attribute__((ext_vector_type(8)));

// ---------------------------------------------------------------------------
// Fused x-projection GEMM: for w in {0,1,3,4}:
//   OUT_w[M x 128] = X[M x 128] @ Wbase[w] + bias[w]
// Block = 128 threads = 4 wave32; block covers 64 rows. The 16x128 A strip is
// staged in LDS ONCE and reused for all four weight matrices (X read 1x).
// W is staged TRANSPOSED (sWt[col][k], even pad) so each B fragment
// {W[k][c], W[k+1][c]} is one aligned ds_load_b64 -> no reg re-pairing movs.
// ---------------------------------------------------------------------------
__global__ __launch_bounds__(128)
void gemm128_x4_kernel(const float* __restrict__ X, const float* __restrict__ Wbase,
                       const float* __restrict__ biasBase,
                       float* __restrict__ o0, float* __restrict__ o1,
                       float* __restrict__ o2, float* __restrict__ o3, int M) {
    __shared__ float sA[4][16][132];   // full-K strip per wave, 33 KB, pad 132 (bank-safe, 8B-aligned)
    __shared__ float sWt[128][34];     // transposed 32-K chunk of W, 17 KB

    const int tid  = threadIdx.x;      // 0..127
    const int wave = tid >> 5;
    const int lane = tid & 31;
    const int half = lane >> 4;
    const int lo   = lane & 15;
    const long row0 = (long)blockIdx.x * 64 + (long)wave * 16;
    const bool rowok = (row0 < (long)M);    // M is a multiple of 16

    // Stage full A strip once (coalesced: one 128B row segment per (i,kp)).
    if (rowok) {
        #pragma unroll
        for (int i = 0; i < 16; ++i) {
            #pragma unroll
            for (int kp = 0; kp < 4; ++kp) {
                sA[wave][i][kp * 32 + lane] = X[(row0 + i) * D + kp * 32 + lane];
            }
        }
    }

    const int   widx[4] = {0, 1, 3, 4};
    float* const outs[4] = {o0, o1, o2, o3};

    for (int w = 0; w < 4; ++w) {
        const float* Wm = Wbase + (long)widx[w] * D * D;
        v8f acc[8] = {};

        for (int kc = 0; kc < 4; ++kc) {
            const int k0 = kc * 32;
            __syncthreads();               // previous chunk's sWt reads done
            // Stage transposed W chunk: sWt[c][r] = W[k0+r][c]; 32 rows x 128 cols.
            #pragma unroll
            for (int i = 0; i < 32; ++i) {
                sWt[tid][i] = Wm[(k0 + i) * D + tid];   // coalesced global read
            }
            __syncthreads();

            #pragma unroll
            for (int ks = 0; ks < 8; ++ks) {
                const int kk = ks * 4 + 2 * half;       // even -> aligned b64
                v2f a;
                a.x = sA[wave][lo][k0 + kk + 0];
                a.y = sA[wave][lo][k0 + kk + 1];
                #pragma unroll
                for (int t = 0; t < 8; ++t) {
                    v2f b;
                    b.x = sWt[t * 16 + lo][kk + 0];
                    b.y = sWt[t * 16 + lo][kk + 1];
                    acc[t] = __builtin_amdgcn_wmma_f32_16x16x4_f32(
                        false, a, false, b, (short)0, acc[t], false, false);
                }
            }
        }

        if (rowok) {
            const float* bias = biasBase + (long)widx[w] * D;
            float* OUT = outs[w];
            #pragma unroll
            for (int t = 0; t < 8; ++t) {
                const int col = t * 16 + lo;
                const float bb = bias[col];
                #pragma unroll
                for (int r = 0; r < 8; ++r) {
                    const long row = row0 + r + 8 * half;
                    OUT[row * D + col] = acc[t][r] + bb;
                }
            }
        }
    }
}

// ---------------------------------------------------------------------------
// Single-matrix GEMM (y-projection): OUT = X @ W + bias.
// Block = 256 threads = 8 waves; 128 rows/block; K chunked by 32; W transposed
// in LDS (same b64-friendly layout as above).
// ---------------------------------------------------------------------------
__global__ __launch_bounds__(256)
void gemm128_kernel(const float* __restrict__ X, const float* __restrict__ W,
                    const float* __restrict__ bias, float* __restrict__ OUT,
                    int M) {
    __shared__ float sWt[128][34];     // transposed 32-K chunk, 17 KB
    __shared__ float sA[8][16][34];    // per-wave 16x32 A tile, 17 KB

    const int tid  = threadIdx.x;
    const int wave = tid >> 5;
    const int lane = tid & 31;
    const int half = lane >> 4;
    const int lo   = lane & 15;
    const long row0 = (long)blockIdx.x * 128 + (long)wave * 16;
    const bool rowok = (row0 < (long)M);

    v8f acc[8] = {};

    for (int kc = 0; kc < 4; ++kc) {
        const int k0 = kc * 32;
        __syncthreads();
        // Stage transposed W chunk: 4096 vals / 256 threads = 16 each.
        #pragma unroll
        for (int i = 0; i < 16; ++i) {
            int f = tid + i * 256;           // 0..4095
            int r = f >> 7, cc = f & 127;
            sWt[cc][r] = W[(k0 + r) * D + cc];
        }
        // Stage per-wave A tile (16 rows x 32 k), coalesced.
        if (rowok) {
            #pragma unroll
            for (int i = 0; i < 16; ++i) {
                sA[wave][i][lane] = X[(row0 + i) * D + (k0 + lane)];
            }
        }
        __syncthreads();

        #pragma unroll
        for (int ks = 0; ks < 8; ++ks) {
            const int kk = ks * 4 + 2 * half;
            v2f a;
            a.x = sA[wave][lo][kk + 0];
            a.y = sA[wave][lo][kk + 1];
            #pragma unroll
            for (int t = 0; t < 8; ++t) {
                v2f b;
                b.x = sWt[t * 16 + lo][kk + 0];
                b.y = sWt[t * 16 + lo][kk + 1];
                acc[t] = __builtin_amdgcn_wmma_f32_16x16x4_f32(
                    false, a, false, b, (short)0, acc[t], false, false);
            }
        }
    }

    if (rowok) {
        #pragma unroll
        for (int t = 0; t < 8; ++t) {
            const int col = t * 16 + lo;
            const float bb = bias[col];
            #pragma unroll
            for (int r = 0; r < 8; ++r) {
                const long row = row0 + r + 8 * half;
                OUT[row * D + col] = acc[t][r] + bb;
            }
        }
    }
}

// ---------------------------------------------------------------------------
// Edge pass: m = m_part + g0[src] + g1[dst]; sigma = sigmoid(m);
// scatter-add num[dst] += g3[src]*sigma, den[dst] += sigma; m written back.
// One wave per edge, 4 features per lane (float4).
// ---------------------------------------------------------------------------
union F4 { float4 v; float f[4]; };

__global__ __launch_bounds__(256)
void edge_pass_kernel(const int* __restrict__ eidx, int E,
                      const float* __restrict__ g0, const float* __restrict__ g1,
                      const float* __restrict__ g3,
                      float* __restrict__ m, float* __restrict__ num,
                      float* __restrict__ den) {
    const int wave = threadIdx.x >> 5;
    const int lane = threadIdx.x & 31;
    const long e = (long)blockIdx.x * 8 + wave;
    if (e >= (long)E) return;
    const long s = (long)eidx[e];
    const long d = (long)eidx[(long)E + e];
    const int c = lane * 4;

    F4 mp, a0, a1, a3, mo;
    mp.v = *(const float4*)(m  + e * D + c);
    a0.v = *(const float4*)(g0 + s * D + c);
    a1.v = *(const float4*)(g1 + d * D + c);
    a3.v = *(const float4*)(g3 + s * D + c);

    #pragma unroll
    for (int i = 0; i < 4; ++i) {
        const float mv  = mp.f[i] + a0.f[i] + a1.f[i];
        const float sig = 1.0f / (1.0f + __expf(-mv));
        const float bh  = a3.f[i] * sig;
        atomicAdd(&num[d * D + c + i], bh);
        atomicAdd(&den[d * D + c + i], sig);
        mo.f[i] = mv;
    }
    *(float4*)(m + e * D + c) = mo.v;
}

// t = t + num / (den + 1e-6), float4 per thread (in place on t)
__global__ __launch_bounds__(256)
void node_combine_kernel(float* __restrict__ t, const float* __restrict__ num,
                         const float* __restrict__ den, long total4) {
    long i = (long)blockIdx.x * 256 + threadIdx.x;
    if (i >= total4) return;
    const long b = i * 4;
    F4 tv, nv, dv;
    tv.v = *(const float4*)(t + b);
    nv.v = *(const float4*)(num + b);
    dv.v = *(const float4*)(den + b);
    #pragma unroll
    for (int k = 0; k < 4; ++k) tv.f[k] += nv.f[k] / (dv.f[k] + 1e-6f);
    *(float4*)(t + b) = tv.v;
}

// Per-column sum & sum-of-squares -> stats[0..127]=sum, stats[128..255]=sumsq
__global__ __launch_bounds__(256)
void colreduce_kernel(const float* __restrict__ X, long M, long rpb,
                      float* __restrict__ stats) {
    __shared__ float ssum[256], ssq[256];
    const int tid = threadIdx.x;
    const int col = tid & 127;
    const int sub = tid >> 7;
    long r  = (long)blockIdx.x * rpb + sub;
    long re = (long)(blockIdx.x + 1) * rpb;
    if (re > M) re = M;
    float s = 0.0f, q = 0.0f;
    for (; r < re; r += 2) {
        float v = X[r * D + col];
        s += v; q += v * v;
    }
    ssum[tid] = s; ssq[tid] = q;
    __syncthreads();
    if (tid < 128) {
        s = ssum[tid] + ssum[tid + 128];
        q = ssq[tid]  + ssq[tid + 128];
        atomicAdd(&stats[col], s);
        atomicAdd(&stats[128 + col], q);
    }
}

// scale = gamma * rsqrt(var + 1e-5); shift = beta - mean*scale
__global__ void bn_finalize_kernel(const float* __restrict__ stats, float inv_count,
                                   const float* __restrict__ gamma,
                                   const float* __restrict__ beta,
                                   float* __restrict__ ss) {
    int c = threadIdx.x;
    float mean = stats[c] * inv_count;
    float var  = stats[128 + c] * inv_count - mean * mean;
    float sc   = gamma[c] * rsqrtf(var + 1e-5f);
    ss[c]       = sc;
    ss[128 + c] = beta[c] - mean * sc;
}

// out = x + silu(scale*t + shift), float4 per thread
__global__ __launch_bounds__(256)
void apply_residual_kernel(const float* __restrict__ x, const float* __restrict__ t,
                           const float* __restrict__ ss, float* __restrict__ out,
                           long total4) {
    long i = (long)blockIdx.x * 256 + threadIdx.x;
    if (i >= total4) return;
    const long b = i * 4;
    const int c = (int)(b & 127);          // multiple of 4, no wrap within float4
    F4 xv, tv, sc, sh, ov;
    xv.v = *(const float4*)(x + b);
    tv.v = *(const float4*)(t + b);
    sc.v = *(const float4*)(ss + c);
    sh.v = *(const float4*)(ss + 128 + c);
    #pragma unroll
    for (int k = 0; k < 4; ++k) {
        const float z = sc.f[k] * tv.f[k] + sh.f[k];
        ov.f[k] = xv.f[k] + z / (1.0f + __expf(-z));
    }
    *(float4*)(out + b) = ov.v;
}

// ---------------------------------------------------------------------------
extern "C" void kernel_launch(void* const* d_in, const int* in_sizes, int n_in,
                              void* d_out, int out_size, void* d_ws, size_t ws_size,
                              hipStream_t stream) {
    (void)in_sizes; (void)n_in; (void)out_size; (void)ws_size;

    const float* atom = (const float*)d_in[0];
    const float* bond = (const float*)d_in[1];
    const float* trip = (const float*)d_in[2];
    const int*   eidx = (const int*)d_in[3];      // [2, NE]
    const int*   aidx = (const int*)d_in[4];      // [2, NA]
    const float* W_ba = (const float*)d_in[5];
    const float* b_ba = (const float*)d_in[6];
    const float* bn_ba = (const float*)d_in[7];
    const float* W_ab = (const float*)d_in[8];
    const float* b_ab = (const float*)d_in[9];
    const float* bn_ab = (const float*)d_in[10];

    float* out_atom = (float*)d_out;                               // [NN,D]
    float* out_bond = out_atom + (long)NN * D;                     // [NE,D]
    float* out_trip = out_bond + (long)NE * D;                     // [NA,D]

    const long SL = (long)NE * D;          // 25.6M floats per big slot
    const long SM = (long)NN * D;          // 2.56M floats (layer-2 tiles)
    float* ws   = (float*)d_ws;
    float* g0   = ws;                      // L1 g0 ; L2 m2
    float* g1   = ws + 1 * SL;             // L1 g1
    float* g3   = ws + 2 * SL;             // L1 g3 ; L2 small tiles live here
    float* tb   = ws + 3 * SL;             // L1 t (g4 -> g4+num/den)
    float* nm   = ws + 4 * SL;             // num accumulator
    float* dn   = ws + 5 * SL;             // den accumulator
    float* st   = ws + 6 * SL;             // stats: 4x256 sums + 4x256 scale/shift
    float* st_n1 = st,        *st_e1 = st + 256, *st_n2 = st + 512, *st_e2 = st + 768;
    float* ss_n1 = st + 1024, *ss_e1 = st + 1280, *ss_n2 = st + 1536, *ss_e2 = st + 1792;
    // Layer-2 small tiles packed into the (then-free) g3 slot:
    float* g0_2 = g3;
    float* g1_2 = g3 + SM;
    float* g3_2 = g3 + 2 * SM;
    float* t2   = g3 + 3 * SM;
    float* m2   = g0;                      // g0 slot free once layer 1 is done

    const int DD = D * D;
    const long rpb = 2048;                 // rows per colreduce block

    // Zero BN accumulators once per call (graph-capture-safe memset nodes).
    hipMemsetAsync(st, 0, 1024 * sizeof(float), stream);

    // =================== Layer 1: bonds (nodes) / angles (edges) ============
    gemm128_x4_kernel<<<dim3((NE + 63) / 64), dim3(128), 0, stream>>>(
        bond, W_ba, b_ba, g0, g1, g3, tb, NE);
    gemm128_kernel<<<dim3((NA + 127) / 128), dim3(256), 0, stream>>>(
        trip, W_ba + 2 * DD, b_ba + 2 * D, out_trip /*m scratch*/, NA);

    hipMemsetAsync(nm, 0, SL * sizeof(float), stream);
    hipMemsetAsync(dn, 0, SL * sizeof(float), stream);

    edge_pass_kernel<<<dim3((NA + 7) / 8), dim3(256), 0, stream>>>(
        aidx, NA, g0, g1, g3, out_trip, nm, dn);
    node_combine_kernel<<<dim3((unsigned)((SL / 4 + 255) / 256)), dim3(256), 0, stream>>>(
        tb, nm, dn, SL / 4);
    colreduce_kernel<<<dim3((unsigned)((NE + rpb - 1) / rpb)), dim3(256), 0, stream>>>(
        tb, NE, rpb, st_n1);
    colreduce_kernel<<<dim3((unsigned)((NA + rpb - 1) / rpb)), dim3(256), 0, stream>>>(
        out_trip, NA, rpb, st_e1);
    bn_finalize_kernel<<<1, 128, 0, stream>>>(st_n1, 1.0f / NE, bn_ba + 0,   bn_ba + 128, ss_n1);
    bn_finalize_kernel<<<1, 128, 0, stream>>>(st_e1, 1.0f / NA, bn_ba + 256, bn_ba + 384, ss_e1);
    apply_residual_kernel<<<dim3((unsigned)((SL / 4 + 255) / 256)), dim3(256), 0, stream>>>(
        bond, tb, ss_n1, out_bond /* bond intermediate */, SL / 4);
    apply_residual_kernel<<<dim3((unsigned)(((long)NA * D / 4 + 255) / 256)), dim3(256), 0, stream>>>(
        trip, out_trip, ss_e1, out_trip, (long)NA * D / 4);

    // =================== Layer 2: atoms (nodes) / bonds (edges) =============
    gemm128_x4_kernel<<<dim3((NN + 63) / 64), dim3(128), 0, stream>>>(
        atom, W_ab, b_ab, g0_2, g1_2, g3_2, t2, NN);
    gemm128_kernel<<<dim3((NE + 127) / 128), dim3(256), 0, stream>>>(
        out_bond /* y = bond intermediate */, W_ab + 2 * DD, b_ab + 2 * D, m2, NE);

    hipMemsetAsync(nm, 0, SM * sizeof(float), stream);
    hipMemsetAsync(dn, 0, SM * sizeof(float), stream);

    edge_pass_kernel<<<dim3((NE + 7) / 8), dim3(256), 0, stream>>>(
        eidx, NE, g0_2, g1_2, g3_2, m2, nm, dn);
    node_combine_kernel<<<dim3((unsigned)((SM / 4 + 255) / 256)), dim3(256), 0, stream>>>(
        t2, nm, dn, SM / 4);
    colreduce_kernel<<<dim3((unsigned)((NN + rpb - 1) / rpb)), dim3(256), 0, stream>>>(
        t2, NN, rpb, st_n2);
    colreduce_kernel<<<dim3((unsigned)((NE + rpb - 1) / rpb)), dim3(256), 0, stream>>>(
        m2, NE, rpb, st_e2);
    bn_finalize_kernel<<<1, 128, 0, stream>>>(st_n2, 1.0f / NN, bn_ab + 0,   bn_ab + 128, ss_n2);
    bn_finalize_kernel<<<1, 128, 0, stream>>>(st_e2, 1.0f / NE, bn_ab + 256, bn_ab + 384, ss_e2);
    apply_residual_kernel<<<dim3((unsigned)((SM / 4 + 255) / 256)), dim3(256), 0, stream>>>(
        atom, t2, ss_n2, out_atom, SM / 4);
    apply_residual_kernel<<<dim3((unsigned)((SL / 4 + 255) / 256)), dim3(256), 0, stream>>>(
        out_bond, m2, ss_e2, out_bond, SL / 4);
}